// FilterPool_59081570124306
// MI455X (gfx1250) — compile-verified
//
#include <hip/hip_runtime.h>
#include <hip/hip_bf16.h>

// ---------------------------------------------------------------------------
// PrRoIPool (FilterPool) for MI455X / gfx1250.
//
// out[n,c,p,q] = (1/area_n) * sum_{h,w} feat[n,c,h,w] * Wy_n[p,h] * Wx_n[q,w]
//
// Strategy (bandwidth-optimal + WMMA):
//  * Only the ROI window (<=29x29 of the 72x72 map) has nonzero weights ->
//    read just that window per (n, channel-chunk): ~6-20x less HBM traffic.
//  * Stage the window for 16 channels in LDS (coalesced global loads).
//  * Flatten (h,w) -> e and run a GEMM: Out[c,pq] = sum_e A[c,e] * B[e,pq]
//    with B[e,pq] = Wy[p,h(e)]*Wx[q,w(e)], via V_WMMA_F32_16X16X4_F32:
//    M=16 channels, N=16 pq columns (2 tiles cover 25 outputs), K stepped by 4.
//  * K tail is padded: LDS feat tail is zero-filled and pad-row weights are
//    exactly 0 (triangle support ends before the window edge), so padding
//    contributes exact zeros (no NaN from 0*garbage).
// ---------------------------------------------------------------------------

typedef __attribute__((ext_vector_type(2))) float v2f;
typedef __attribute__((ext_vector_type(8))) float v8f;

#define K_POOL   5
#define SCALE_F  (1.0f / 16.0f)
#define NCH      16      // channels per workgroup
#define THREADS  128     // 4 waves; waves 0/1 do WMMA, all 4 load
#define MAXWIN   848     // >= padded max window (29*29=841 -> 844), margin
#define WPAD     80      // padded row length for weight tables (H=W=72 -> 75 max index)

__device__ __forceinline__ float tri_int(float t) {
    // Antiderivative of max(0, 1-|t|), clipped to support.
    t = fminf(1.0f, fmaxf(-1.0f, t));
    return (t < 0.0f) ? 0.5f * (t + 1.0f) * (t + 1.0f)
                      : 0.5f + t - 0.5f * t * t;
}

__global__ void __launch_bounds__(THREADS)
filterpool_wmma_kernel(const float* __restrict__ feat,
                       const float* __restrict__ bb,
                       float* __restrict__ out,
                       int N, int C, int H, int W)
{
    __shared__ float wyx[2][K_POOL][WPAD];       // [0]=Wy over h, [1]=Wx over w
    __shared__ float featLds[NCH * MAXWIN];      // window, per-channel stride winPad
    __shared__ unsigned hwLds[MAXWIN];           // packed (h<<16)|w per window elem

    const int n   = blockIdx.y;
    const int c0  = blockIdx.x * NCH;
    const int tid = threadIdx.x;

    // ---- box parameters (uniform: every thread recomputes, loads broadcast) ----
    const float bx = bb[4 * n + 0], by = bb[4 * n + 1];
    const float bw = bb[4 * n + 2], bh = bb[4 * n + 3];
    const float x1 = bx * SCALE_F,        y1 = by * SCALE_F;
    const float x2 = (bx + bw) * SCALE_F, y2 = (by + bh) * SCALE_F;
    const float binw = (x2 - x1) / K_POOL, binh = (y2 - y1) / K_POOL;

    // nonzero-weight window: pixel i contributes iff i > lo-1 && i < hi+1
    int w_lo = max(0, (int)floorf(x1) - 1);
    int w_hi = min(W, (int)ceilf(x2) + 1);
    int h_lo = max(0, (int)floorf(y1) - 1);
    int h_hi = min(H, (int)ceilf(y2) + 1);
    int WW = max(0, w_hi - w_lo);
    int WH = max(0, h_hi - h_lo);
    int win = WW * WH;
    if (win > MAXWIN - 4) win = MAXWIN - 4;      // safety clamp (never hit for valid boxes)
    const int winPad = (win + 3) & ~3;           // K padded to multiple of 4

    // ---- weight tables Wy (ax=0) / Wx (ax=1) into LDS ----
    for (int idx = tid; idx < 2 * K_POOL * WPAD; idx += THREADS) {
        int ax  = idx / (K_POOL * WPAD);
        int rem = idx - ax * (K_POOL * WPAD);
        int p   = rem / WPAD;
        int i   = rem - p * WPAD;
        float lo  = ax ? x1 : y1;
        float bs  = ax ? binw : binh;
        int   lim = ax ? W : H;
        float val = 0.0f;
        if (i < lim) {
            float e0 = lo + bs * (float)p;
            float e1 = lo + bs * (float)(p + 1);
            val = tri_int(e1 - (float)i) - tri_int(e0 - (float)i);
        }
        wyx[ax][p][i] = val;
    }

    // ---- e -> (h,w) map (covers the zero-weight pad rows too) ----
    const int wwSafe = max(WW, 1);
    for (int e = tid; e < winPad; e += THREADS) {
        int hh = e / wwSafe;
        int ww = e - hh * wwSafe;
        hwLds[e] = ((unsigned)(h_lo + hh) << 16) | (unsigned)(w_lo + ww);
    }

    // ---- stream the feat window for NCH channels into LDS (zero tail) ----
    const float* fbase = feat + ((size_t)n * C + c0) * (size_t)(H * W);
    for (int idx = tid; idx < NCH * winPad; idx += THREADS) {
        int ch = idx / winPad;
        int e  = idx - ch * winPad;
        float v = 0.0f;
        if (e < win) {
            int hh = e / WW;                      // WW>0 whenever win>0
            int ww = e - hh * WW;
            v = fbase[(size_t)ch * (H * W) + (size_t)(h_lo + hh) * W + (w_lo + ww)];
        }
        featLds[ch * winPad + e] = v;
    }
    __syncthreads();

    // ---- WMMA GEMM: waves 0/1 own pq tiles 0/1 (branch is wave-uniform -> EXEC all 1s) ----
    const int wave = tid >> 5;
    const int lane = tid & 31;
    if (wave < 2) {
        const int  col      = lane & 15;          // M index for A, N index for B/D
        const int  hi       = lane >> 4;          // half-wave: K+0/K+1 vs K+2/K+3
        const int  pq       = wave * 16 + col;
        const bool colValid = (pq < K_POOL * K_POOL);
        const int  p        = min(pq / K_POOL, K_POOL - 1);
        const int  q        = min(pq - K_POOL * (pq / K_POOL), K_POOL - 1);

        v8f acc = {0.f, 0.f, 0.f, 0.f, 0.f, 0.f, 0.f, 0.f};

        for (int k = 0; k < winPad; k += 4) {
            const int r0 = k + hi * 2;            // this half-wave's K rows: r0, r0+1
            // A (16x4 f32): lanes 0-15 M=0..15 hold K=k+0 (v0) / k+1 (v1);
            //               lanes 16-31 hold K=k+2 / k+3.
            v2f a;
            a.x = featLds[col * winPad + r0 + 0];
            a.y = featLds[col * winPad + r0 + 1];
            // B (4x16 f32): one row per half-VGPR, N striped across lanes.
            v2f b;
            if (colValid) {
                unsigned hw0 = hwLds[r0 + 0];
                unsigned hw1 = hwLds[r0 + 1];
                b.x = wyx[0][p][hw0 >> 16] * wyx[1][q][hw0 & 0xffffu];
                b.y = wyx[0][p][hw1 >> 16] * wyx[1][q][hw1 & 0xffffu];
            } else {
                b.x = 0.f; b.y = 0.f;
            }
            // D = A*B + C  (f32 accumulate)
            acc = __builtin_amdgcn_wmma_f32_16x16x4_f32(
                    /*neg_a=*/false, a, /*neg_b=*/false, b,
                    /*c_mod=*/(short)0, acc,
                    /*reuse_a=*/false, /*reuse_b=*/false);
        }

        // ---- scale by 1/area and store (matches jnp.where(area>0, integ/safe, 0)) ----
        const float area = fmaxf(binw, 0.0f) * fmaxf(binh, 0.0f);
        const float inv  = (area > 0.0f) ? (1.0f / fmaxf(area, 1e-12f)) : 0.0f;
        if (colValid) {
            // D layout: VGPR r -> M=r (lanes 0-15) or M=r+8 (lanes 16-31), N=col
#pragma unroll
            for (int r = 0; r < 8; ++r) {
                int ch = c0 + r + hi * 8;
                out[((size_t)n * C + ch) * (K_POOL * K_POOL) + pq] = acc[r] * inv;
            }
        }
    }
}

extern "C" void kernel_launch(void* const* d_in, const int* in_sizes, int n_in,
                              void* d_out, int out_size, void* d_ws, size_t ws_size,
                              hipStream_t stream) {
    const float* feat = (const float*)d_in[0];   // [64,256,72,72] f32
    const float* bb   = (const float*)d_in[1];   // [64,4] f32
    float*       out  = (float*)d_out;           // [64,256,5,5] f32

    const int N = 64, C = 256, H = 72, W = 72;
    dim3 grid(C / NCH, N);                       // 16 x 64 = 1024 workgroups
    filterpool_wmma_kernel<<<grid, THREADS, 0, stream>>>(feat, bb, out, N, C, H, W);
}